// ModelNew_73315091744326
// MI455X (gfx1250) — compile-verified
//
#include <hip/hip_runtime.h>

#define TN 4096
#define BM 64
#define BN 64
#define BK 32
#define LDA 36  // padded LDS stride for A (floats); rows stay 16B-aligned

typedef float v2f __attribute__((ext_vector_type(2)));
typedef float v4f __attribute__((ext_vector_type(4)));
typedef float v8f __attribute__((ext_vector_type(8)));

// ---- global -> registers (masked on triangular edge blocks only) ----
__device__ __forceinline__ void gload_tile(const float* __restrict__ A,
                                           const float* __restrict__ B,
                                           int bi, int bj, int kb, int tid,
                                           v4f ra[4], v4f rb[2][2]) {
  const int k0 = kb * BK;
  const bool mA = (kb <= 2 * bi + 1);  // A[i,k]=0 for k<i possible only here
  const bool mB = (kb >= 2 * bj);      // B[k,j]=0 for k>j possible only here
#pragma unroll
  for (int r = 0; r < 4; ++r) {
    int idx = tid + r * 128;            // 0..511 float4 slots of A tile
    int row = idx >> 3;                 // 0..63
    int kc  = (idx & 7) << 2;           // 0..28
    int gi  = bi * BM + row;
    int gk  = k0 + kc;
    v4f v = *(const v4f*)(A + gi * TN + gk);
    if (mA) {
      v.x = (gk + 0 >= gi) ? v.x : 0.f;
      v.y = (gk + 1 >= gi) ? v.y : 0.f;
      v.z = (gk + 2 >= gi) ? v.z : 0.f;
      v.w = (gk + 3 >= gi) ? v.w : 0.f;
    }
    ra[r] = v;
  }
#pragma unroll
  for (int r = 0; r < 2; ++r) {
    int idx = tid + r * 128;            // 0..255 pair-slots of B tile
    int p   = idx >> 4;                 // k-pair index 0..15
    int nc  = (idx & 15) << 2;          // 0..60
    int gk  = k0 + 2 * p;
    int gj  = bj * BN + nc;
    v4f v0 = *(const v4f*)(B + gk * TN + gj);
    v4f v1 = *(const v4f*)(B + (gk + 1) * TN + gj);
    if (mB) {
      v0.x = (gk     <= gj + 0) ? v0.x : 0.f;
      v0.y = (gk     <= gj + 1) ? v0.y : 0.f;
      v0.z = (gk     <= gj + 2) ? v0.z : 0.f;
      v0.w = (gk     <= gj + 3) ? v0.w : 0.f;
      v1.x = (gk + 1 <= gj + 0) ? v1.x : 0.f;
      v1.y = (gk + 1 <= gj + 1) ? v1.y : 0.f;
      v1.z = (gk + 1 <= gj + 2) ? v1.z : 0.f;
      v1.w = (gk + 1 <= gj + 3) ? v1.w : 0.f;
    }
    rb[r][0] = v0;
    rb[r][1] = v1;
  }
}

// ---- registers -> LDS; B stored k-pair interleaved: sB[(p*BN + n)*2 + (k&1)] ----
__device__ __forceinline__ void lds_store_tile(float* sA, float* sB, int tid,
                                               const v4f ra[4], const v4f rb[2][2]) {
#pragma unroll
  for (int r = 0; r < 4; ++r) {
    int idx = tid + r * 128;
    int row = idx >> 3;
    int kc  = (idx & 7) << 2;
    *(v4f*)(sA + row * LDA + kc) = ra[r];
  }
#pragma unroll
  for (int r = 0; r < 2; ++r) {
    int idx = tid + r * 128;
    int p   = idx >> 4;
    int nc  = (idx & 15) << 2;
#pragma unroll
    for (int t = 0; t < 4; ++t) {
      v2f pr;
      pr.x = rb[r][0][t];   // B[2p  ][nc+t]
      pr.y = rb[r][1][t];   // B[2p+1][nc+t]
      *(v2f*)(sB + (p * BN + nc + t) * 2) = pr;
    }
  }
}

__global__ __launch_bounds__(128)
void triu_gemm_wmma_f32(const float* __restrict__ A,
                        const float* __restrict__ B,
                        float* __restrict__ C) {
  const int bj  = blockIdx.x;
  const int bi  = blockIdx.y;
  const int tid = threadIdx.x;

  // Lower-triangular tiles are identically zero (uniform branch, EXEC stays full).
  if (bj < bi) {
    const int base = (bi * BM) * TN + bj * BN;
#pragma unroll
    for (int r = 0; r < 8; ++r) {
      int idx = tid + r * 128;
      int row = idx >> 4;
      int c   = (idx & 15) << 2;
      v4f z = {0.f, 0.f, 0.f, 0.f};
      *(v4f*)(C + base + row * TN + c) = z;
    }
    return;
  }

  __shared__ float sA[2][BM * LDA];   // ping-pong A tiles
  __shared__ float sB[2][BK * BN];    // ping-pong B tiles (k-pair interleaved)

  const int lane = tid & 31;
  const int wave = tid >> 5;
  const int m0   = wave * 16;          // wave's 16-row strip
  const int fm   = lane & 15;          // M for A frag / N for B frag & C cols
  const int kh   = (lane >> 4) << 1;   // 0 (lanes 0-15) or 2 (lanes 16-31)

  v8f acc0 = {}, acc1 = {}, acc2 = {}, acc3 = {};

  const int kb_first = 2 * bi;         // k-blocks of width BK=32 covering [64*bi, 64*(bj+1))
  const int kb_last  = 2 * bj + 1;

  v4f ra[4];
  v4f rb[2][2];

  // prologue: fill buffer 0, prefetch block kb_first+1 into registers
  gload_tile(A, B, bi, bj, kb_first, tid, ra, rb);
  lds_store_tile(sA[0], sB[0], tid, ra, rb);
  gload_tile(A, B, bi, bj, kb_first + 1, tid, ra, rb);
  __syncthreads();

  for (int kb = kb_first; kb <= kb_last; ++kb) {
    const int cur = (kb - kb_first) & 1;
    const float* cA = sA[cur];
    const float* cB = sB[cur];

#pragma unroll
    for (int kk = 0; kk < BK; kk += 4) {
      // A fragment 16x4: lanes 0-15 hold K=kk,kk+1; lanes 16-31 hold K=kk+2,kk+3
      v2f a = *(const v2f*)(cA + (m0 + fm) * LDA + kk + kh);
      // B fragments 4x16: interleaved pairs -> single b64 load each
      const int pb = ((kk + kh) >> 1) * (BN * 2) + fm * 2;
      v2f b0 = *(const v2f*)(cB + pb + 0 * 32);
      v2f b1 = *(const v2f*)(cB + pb + 1 * 32);
      v2f b2 = *(const v2f*)(cB + pb + 2 * 32);
      v2f b3 = *(const v2f*)(cB + pb + 3 * 32);

      acc0 = __builtin_amdgcn_wmma_f32_16x16x4_f32(false, a, false, b0, (short)0, acc0, false, false);
      acc1 = __builtin_amdgcn_wmma_f32_16x16x4_f32(false, a, false, b1, (short)0, acc1, false, false);
      acc2 = __builtin_amdgcn_wmma_f32_16x16x4_f32(false, a, false, b2, (short)0, acc2, false, false);
      acc3 = __builtin_amdgcn_wmma_f32_16x16x4_f32(false, a, false, b3, (short)0, acc3, false, false);
    }

    if (kb < kb_last) {
      __syncthreads();                               // everyone done reading sA/sB[cur^1] last iter
      lds_store_tile(sA[cur ^ 1], sB[cur ^ 1], tid, ra, rb);
      if (kb + 1 < kb_last)
        gload_tile(A, B, bi, bj, kb + 2, tid, ra, rb);  // hide VMEM under next compute
      __syncthreads();                               // stores visible before next compute
    }
  }

  // store C: D layout VGPR r -> M = r + 8*(lane>=16), N = lane&15 (coalesced b32)
  const int half = lane >> 4;
  const int ccol = bj * BN + fm;
#pragma unroll
  for (int r = 0; r < 8; ++r) {
    int crow = bi * BM + m0 + r + 8 * half;
    float* cp = C + crow * TN + ccol;
    cp[0]  = acc0[r];
    cp[16] = acc1[r];
    cp[32] = acc2[r];
    cp[48] = acc3[r];
  }
}

extern "C" void kernel_launch(void* const* d_in, const int* in_sizes, int n_in,
                              void* d_out, int out_size, void* d_ws, size_t ws_size,
                              hipStream_t stream) {
  const float* A = (const float*)d_in[0];
  const float* B = (const float*)d_in[1];
  float* C = (float*)d_out;
  (void)in_sizes; (void)n_in; (void)out_size; (void)d_ws; (void)ws_size;

  dim3 grid(TN / BN, TN / BM);  // (bj, bi) = 64 x 64 tiles
  dim3 block(128);
  triu_gemm_wmma_f32<<<grid, block, 0, stream>>>(A, B, C);
}